// MultiVariateKL_divergence_11467562680393
// MI455X (gfx1250) — compile-verified
//
#include <hip/hip_runtime.h>
#include <hip/hip_bf16.h>

// ---------- types ----------
typedef float v2f __attribute__((ext_vector_type(2)));
typedef float v4f __attribute__((ext_vector_type(4)));
typedef float v8f __attribute__((ext_vector_type(8)));

#define TPB 256
#define NBLK 2048

// ds_swizzle_b32 lane-XOR add (Group-of-32 encoding: and=0x1F, or=0, xor=mask).
// Single DS instruction instead of the generic __shfl_xor bpermute sequence.
#define SWZ_XOR_ADD(s, imm_off)                                              \
    do {                                                                     \
        int _t = __builtin_amdgcn_ds_swizzle(__float_as_int(s), (imm_off));  \
        (s) += __int_as_float(_t);                                           \
    } while (0)

#define SWZ_X16 0x401F  /* xor 16 */
#define SWZ_X4  0x101F  /* xor 4  */
#define SWZ_X2  0x081F  /* xor 2  */
#define SWZ_X1  0x041F  /* xor 1  */

// Sum a per-lane float across the whole wave32 using V_WMMA_F32_16X16X4_F32.
// A (16x4, documented layout): VGPR0 lane m -> A[m][0] (lanes 0-15), A[m][2]
// (lanes 16-31); VGPR1 = 0. B = all-ones (layout-independent since uniform).
// D[m][n] = v_m + v_{m+16} for every column n; lane L holds D[M=r(+8)][N=L%16]
// in VGPRs r=0..7, so summing the 8 D regs + cross-half swap (xor 16) yields
// the full wave sum in every lane. Callers must be in wave-uniform control
// flow (EXEC all 1s for the executing wave).
__device__ __forceinline__ float wave_sum_wmma(float v) {
    v2f a; a[0] = v;    a[1] = 0.0f;
    v2f b; b[0] = 1.0f; b[1] = 1.0f;
    v8f c = {};
    c = __builtin_amdgcn_wmma_f32_16x16x4_f32(
            /*neg_a=*/false, a, /*neg_b=*/false, b,
            /*c_mod=*/(short)0, c, /*reuse_a=*/false, /*reuse_b=*/false);
    float s = ((c[0] + c[1]) + (c[2] + c[3])) + ((c[4] + c[5]) + (c[6] + c[7]));
    SWZ_XOR_ADD(s, SWZ_X16);
    return s;
}

__device__ __forceinline__ float kl_elem(float mp, float lsp, float mq, float lsq) {
    const float LOG2E = 1.44269504088896340736f;
    float dl   = lsq - lsp;
    float e1   = __builtin_amdgcn_exp2f(-dl * LOG2E);   // exp(lsp - lsq)
    float e2   = __builtin_amdgcn_exp2f(-lsq * LOG2E);  // exp(-lsq)
    float diff = mq - mp;
    return fmaf(diff * diff, e2, dl + e1);
}

__device__ __forceinline__ float kl_vec4(const v4f& a, const v4f& b,
                                         const v4f& c, const v4f& d) {
    float s0 = kl_elem(a[0], b[0], c[0], d[0]);
    float s1 = kl_elem(a[1], b[1], c[1], d[1]);
    float s2 = kl_elem(a[2], b[2], c[2], d[2]);
    float s3 = kl_elem(a[3], b[3], c[3], d[3]);
    return (s0 + s1) + (s2 + s3);
}

// Stage 1: streaming partial reduction. One partial per block -> ws[blockIdx.x].
__global__ __launch_bounds__(TPB)
void kl_partial(const float* __restrict__ mp,  const float* __restrict__ lsp,
                const float* __restrict__ mq,  const float* __restrict__ lsq,
                float* __restrict__ ws, long long n4, long long n) {
    const long long tid    = (long long)blockIdx.x * blockDim.x + threadIdx.x;
    const long long stride = (long long)gridDim.x * blockDim.x;

    const v4f* __restrict__ mp4  = (const v4f*)mp;
    const v4f* __restrict__ lsp4 = (const v4f*)lsp;
    const v4f* __restrict__ mq4  = (const v4f*)mq;
    const v4f* __restrict__ lsq4 = (const v4f*)lsq;

    float acc = 0.0f;
    long long i = tid;
    // 2x unrolled: 8 outstanding global_load_b128 (TH=NT) before first use.
    for (; i + stride < n4; i += 2 * stride) {
        const long long j = i + stride;
        v4f a0 = __builtin_nontemporal_load(mp4  + i);
        v4f b0 = __builtin_nontemporal_load(lsp4 + i);
        v4f c0 = __builtin_nontemporal_load(mq4  + i);
        v4f d0 = __builtin_nontemporal_load(lsq4 + i);
        v4f a1 = __builtin_nontemporal_load(mp4  + j);
        v4f b1 = __builtin_nontemporal_load(lsp4 + j);
        v4f c1 = __builtin_nontemporal_load(mq4  + j);
        v4f d1 = __builtin_nontemporal_load(lsq4 + j);
        acc += kl_vec4(a0, b0, c0, d0);
        acc += kl_vec4(a1, b1, c1, d1);
    }
    for (; i < n4; i += stride) {
        v4f a = __builtin_nontemporal_load(mp4  + i);
        v4f b = __builtin_nontemporal_load(lsp4 + i);
        v4f c = __builtin_nontemporal_load(mq4  + i);
        v4f d = __builtin_nontemporal_load(lsq4 + i);
        acc += kl_vec4(a, b, c, d);
    }
    // Scalar tail (n divisible by 4 for this problem; zero iterations).
    if (blockIdx.x == 0 && threadIdx.x == 0) {
        for (long long k = n4 * 4; k < n; ++k)
            acc += kl_elem(mp[k], lsp[k], mq[k], lsq[k]);
    }

    // Wave reduction via WMMA (full EXEC here), then LDS across the 8 waves.
    float wsum = wave_sum_wmma(acc);

    __shared__ float slds[TPB / 32];
    const int lane = threadIdx.x & 31;
    const int wave = threadIdx.x >> 5;
    if (lane == 0) slds[wave] = wsum;
    __syncthreads();

    if (threadIdx.x < 32) {
        float v = (lane < TPB / 32) ? slds[lane] : 0.0f;
        SWZ_XOR_ADD(v, SWZ_X4);
        SWZ_XOR_ADD(v, SWZ_X2);
        SWZ_XOR_ADD(v, SWZ_X1);
        if (lane == 0) ws[blockIdx.x] = v;
    }
}

// Stage 2: single block folds the NBLK partials and writes the scalar result.
__global__ __launch_bounds__(TPB)
void kl_final(const float* __restrict__ ws, int nblk,
              float* __restrict__ out, float inv_nsamples, float dch) {
    float acc = 0.0f;
    for (int i = threadIdx.x; i < nblk; i += TPB) acc += ws[i];

    float wsum = wave_sum_wmma(acc);

    __shared__ float slds[TPB / 32];
    const int lane = threadIdx.x & 31;
    const int wave = threadIdx.x >> 5;
    if (lane == 0) slds[wave] = wsum;
    __syncthreads();

    if (threadIdx.x < 32) {
        float v = (lane < TPB / 32) ? slds[lane] : 0.0f;
        SWZ_XOR_ADD(v, SWZ_X4);
        SWZ_XOR_ADD(v, SWZ_X2);
        SWZ_XOR_ADD(v, SWZ_X1);
        if (lane == 0) out[0] = 0.5f * (v * inv_nsamples - dch);
    }
}

extern "C" void kernel_launch(void* const* d_in, const int* in_sizes, int n_in,
                              void* d_out, int out_size, void* d_ws, size_t ws_size,
                              hipStream_t stream) {
    const float* mean_p      = (const float*)d_in[0];
    const float* log_sigma_p = (const float*)d_in[1];
    const float* mean_q      = (const float*)d_in[2];
    const float* log_sigma_q = (const float*)d_in[3];
    float* out = (float*)d_out;
    float* ws  = (float*)d_ws;

    const long long n   = (long long)in_sizes[0];  // 29,491,200
    const long long n4  = n / 4;
    const int       dch = 3;                       // channel dim of (2,3,160,192,160)
    const double    nsamples = (double)n / (double)dch;

    kl_partial<<<NBLK, TPB, 0, stream>>>(mean_p, log_sigma_p, mean_q, log_sigma_q,
                                         ws, n4, n);
    kl_final<<<1, TPB, 0, stream>>>(ws, NBLK, out,
                                    (float)(1.0 / nsamples), (float)dch);
}